// TransformedSegmentation_84971632984244
// MI455X (gfx1250) — compile-verified
//
#include <hip/hip_runtime.h>

// Problem constants (match reference: B=32, C=3, H=W=512, Hc=Wc=256)
#define BATCH   32
#define Hdim    512
#define Wdim    512
#define HCc     256
#define WCc     256
#define HWPLANE (Hdim * Wdim)          // 262144 elements per image plane
#define NTOT    (BATCH * HWPLANE)      // 8,388,608 elements per output tensor
#define ELEMS_PER_BLOCK 1024           // 256 threads * 4 floats
#define BLOCKS_PER_IMG  (HWPLANE / ELEMS_PER_BLOCK)  // 256

// Native clang vector => legal for __builtin_nontemporal_* and lowers to
// global_load_b128 / global_store_b128 with nontemporal hints.
typedef float v4f __attribute__((ext_vector_type(4)));

// Fused kernel:
//  - copies y_initial -> out[0 .. NTOT)            (pure streaming copy)
//  - computes nearest-paste y -> out[NTOT .. 2*NTOT)
// Each block handles 1024 contiguous elements of exactly one image, so the
// batch index (and therefore the bbox) is wave-uniform -> scalar loads.
__global__ __launch_bounds__(256) void
TransformedSegmentation_paste_kernel(const float* __restrict__ y_initial,
                                     const float* __restrict__ y_cropped,
                                     const int*   __restrict__ bboxes,
                                     float*       __restrict__ out)
{
    const int b       = (int)(blockIdx.x >> 8);                       // blockIdx / 256
    const int img_off = (int)(((blockIdx.x & 255u) << 10) + (threadIdx.x << 2));
    const int gidx    = b * HWPLANE + img_off;                        // multiple of 4

    // ---------- pass-through copy of y_initial (streamed, nontemporal) ----------
    const v4f yi = __builtin_nontemporal_load(
        reinterpret_cast<const v4f*>(y_initial + gidx));
    __builtin_nontemporal_store(yi, reinterpret_cast<v4f*>(out + gidx));

    // ---------- bbox is uniform per block -> compiler emits s_load ----------
    const int x1 = bboxes[b * 4 + 0];
    const int y1 = bboxes[b * 4 + 1];
    const int x2 = bboxes[b * 4 + 2];
    const int y2 = bboxes[b * 4 + 3];
    const unsigned w = (unsigned)(x2 - x1);   // in [128, 256]
    const unsigned h = (unsigned)(y2 - y1);   // in [128, 256]

    const int r  = img_off >> 9;    // row within image (W = 512)
    const int c0 = img_off & 511;   // first of 4 consecutive columns (same row)

    v4f v = (v4f){0.f, 0.f, 0.f, 0.f};
    const bool in_r = (r >= y1) && (r < y2);
    if (in_r) {
        // exact floor division, numerator >= 0 inside the row guard
        unsigned sr = ((unsigned)(r - y1) * (unsigned)HCc) / h;
        if (sr > (HCc - 1)) sr = HCc - 1;   // matches jnp.clip (defensive)
        const float* __restrict__ row = y_cropped + b * (HCc * WCc) + (int)sr * WCc;

#pragma unroll
        for (int j = 0; j < 4; ++j) {
            const int c = c0 + j;
            float val = 0.f;
            if ((c >= x1) && (c < x2)) {
                unsigned sc = ((unsigned)(c - x1) * (unsigned)WCc) / w;
                if (sc > (WCc - 1)) sc = WCc - 1;
                val = row[sc];                 // cached gather; y_cropped fits in L2
            }
            v[j] = val;
        }
    }
    // outside bbox rows: v stays zero -> matches mask semantics
    __builtin_nontemporal_store(v, reinterpret_cast<v4f*>(out + NTOT + gidx));
}

extern "C" void kernel_launch(void* const* d_in, const int* in_sizes, int n_in,
                              void* d_out, int out_size, void* d_ws, size_t ws_size,
                              hipStream_t stream)
{
    // setup_inputs order: x (unused), y_initial, y_cropped, bboxes
    const float* y_initial = (const float*)d_in[1];
    const float* y_cropped = (const float*)d_in[2];
    const int*   bboxes    = (const int*)d_in[3];
    float*       out       = (float*)d_out;

    dim3 grid(BATCH * BLOCKS_PER_IMG);   // 8192 blocks
    dim3 block(256);                     // 8 waves (wave32) per block
    TransformedSegmentation_paste_kernel<<<grid, block, 0, stream>>>(
        y_initial, y_cropped, bboxes, out);
}